// GraphEncoder_43843026157858
// MI455X (gfx1250) — compile-verified
//
#include <hip/hip_runtime.h>

#define N_NODES 100000
#define N_EDGES 1600000
#define DIM_IN  128
#define DIM_HID 128
#define DIM_OUT 64

typedef __attribute__((ext_vector_type(2))) float v2f;
typedef __attribute__((ext_vector_type(8))) float v8f;

// ---------------------------------------------------------------- zero fill
__global__ __launch_bounds__(256) void zero_f32(float* __restrict__ p, size_t n) {
  size_t i = (size_t)blockIdx.x * blockDim.x + threadIdx.x;
  size_t stride = (size_t)gridDim.x * blockDim.x;
  for (; i < n; i += stride) p[i] = 0.0f;
}

// ------------------------------------------------ edge scatter (mean-agg numerator)
// one wave32 per edge: 128 floats = float4 per lane; atomic f32 adds land in L2.
__global__ __launch_bounds__(256) void scatter_edges(
    const float* __restrict__ feat, const int* __restrict__ src,
    const int* __restrict__ dst, float* __restrict__ agg,
    float* __restrict__ deg, int nEdges) {
  int wave = threadIdx.x >> 5;
  int lane = threadIdx.x & 31;
  int edge = blockIdx.x * 8 + wave;
  if (edge >= nEdges) return;
  int s = src[edge];
  int d = dst[edge];
  if ((unsigned)s >= (unsigned)N_NODES || (unsigned)d >= (unsigned)N_NODES) return;
  float4 v = ((const float4*)(feat + (size_t)s * DIM_IN))[lane];
  float* o = agg + (size_t)d * DIM_IN + lane * 4;
  atomicAdd(o + 0, v.x);
  atomicAdd(o + 1, v.y);
  atomicAdd(o + 2, v.z);
  atomicAdd(o + 3, v.w);
  if (deg != nullptr && lane == 0) atomicAdd(deg + d, 1.0f);
}

// ------------------------------------------------ fused SAGE GEMM via f32 WMMA
// out[M,N] = (agg/max(deg,1)) @ Wl + bl + xin @ Wr
// One wave per 16x16 output tile; K accumulated with V_WMMA_F32_16X16X4_F32.
// K, N, NCT are compile-time so the K-loop has a static trip count (no scalar
// guard/peel machinery around the wmma pipeline).
template <int K, int N, int NCT>
__global__ __launch_bounds__(256) void sage_gemm(
    const float* __restrict__ agg, const float* __restrict__ deg,
    const float* __restrict__ xin,
    const float* __restrict__ Wl, const float* __restrict__ bl,
    const float* __restrict__ Wr, float* __restrict__ out) {
  int wave = threadIdx.x >> 5;
  int lane = threadIdx.x & 31;
  constexpr int RTPB = 8 / NCT;                  // row tiles per block (8 waves)
  int rowTile = blockIdx.x * RTPB + wave / NCT;
  int colTile = wave % NCT;

  int half = lane >> 4;                          // 0: K+{0,1}, 1: K+{2,3}
  int h2   = half << 1;
  int l16  = lane & 15;
  int arow = rowTile * 16 + l16;                 // A-fragment row for this lane
  int col  = colTile * 16 + l16;                 // B/C/D column for this lane

  float dg    = deg[arow];
  float scale = 1.0f / fmaxf(dg, 1.0f);          // mean = agg / max(deg,1)
  const float* aAgg = agg + (size_t)arow * K + h2;
  const float* aX   = xin + (size_t)arow * K + h2;
  const float* wl   = Wl + (size_t)h2 * N + col;
  const float* wr   = Wr + (size_t)h2 * N + col;

  float bias = bl[col];
  v8f c;
#pragma unroll
  for (int i = 0; i < 8; ++i) c[i] = bias;

#pragma unroll 4
  for (int k0 = 0; k0 < K; k0 += 4) {
    // A fragment: VGPR0 = A[row][ka], VGPR1 = A[row][ka+1]   (ka = k0 + h2)
    v2f a0 = *(const v2f*)aAgg;
    a0.x *= scale;
    a0.y *= scale;
    // B fragment: VGPR0 = W[ka][col], VGPR1 = W[ka+1][col]
    v2f b0;
    b0.x = wl[0];
    b0.y = wl[N];
    c = __builtin_amdgcn_wmma_f32_16x16x4_f32(false, a0, false, b0, (short)0, c,
                                              false, false);
    v2f a1 = *(const v2f*)aX;
    v2f b1;
    b1.x = wr[0];
    b1.y = wr[N];
    c = __builtin_amdgcn_wmma_f32_16x16x4_f32(false, a1, false, b1, (short)0, c,
                                              false, false);
    aAgg += 4;
    aX += 4;
    wl += 4 * N;
    wr += 4 * N;
  }

  // C/D layout: VGPR v -> row (8*half + v), lane&15 -> col
  int srow = rowTile * 16 + (half << 3);
#pragma unroll
  for (int v = 0; v < 8; ++v) out[(size_t)(srow + v) * N + col] = c[v];
}

// ------------------------------------------------ column sums for PairNorm
__global__ __launch_bounds__(256) void col_sum(const float* __restrict__ h,
                                               float* __restrict__ colsum,
                                               int nRows) {
  int col = threadIdx.x & 127;
  int r = blockIdx.x * 2 + (threadIdx.x >> 7);
  int stride = gridDim.x * 2;
  float s = 0.0f;
  for (; r < nRows; r += stride) s += h[(size_t)r * DIM_HID + col];
  atomicAdd(colsum + col, s);
}

// ------------------------------------------------ PairNorm-SCS + ReLU (in place)
__global__ __launch_bounds__(256) void pairnorm_relu(float* __restrict__ h,
                                                     const float* __restrict__ colsum,
                                                     int nRows) {
  int wave = threadIdx.x >> 5;
  int lane = threadIdx.x & 31;
  int row = blockIdx.x * 8 + wave;
  if (row >= nRows) return;
  float4* p = (float4*)(h + (size_t)row * DIM_HID);
  float4 v = p[lane];
  float ss = v.x * v.x + v.y * v.y + v.z * v.z + v.w * v.w;
#pragma unroll
  for (int off = 16; off > 0; off >>= 1) ss += __shfl_xor(ss, off, 32);
  float inv = 1.0f / sqrtf(1e-6f + ss);          // NORM_SCALE == 1.0
  float4 cm = ((const float4*)colsum)[lane];
  const float invN = 1.0f / (float)N_NODES;
  float4 o;
  o.x = fmaxf(v.x * inv - cm.x * invN, 0.0f);
  o.y = fmaxf(v.y * inv - cm.y * invN, 0.0f);
  o.z = fmaxf(v.z * inv - cm.z * invN, 0.0f);
  o.w = fmaxf(v.w * inv - cm.w * invN, 0.0f);
  p[lane] = o;
}

// ---------------------------------------------------------------- launcher
extern "C" void kernel_launch(void* const* d_in, const int* in_sizes, int n_in,
                              void* d_out, int out_size, void* d_ws, size_t ws_size,
                              hipStream_t stream) {
  (void)in_sizes; (void)n_in; (void)out_size; (void)ws_size;
  const float* x   = (const float*)d_in[0];
  const int*   ei  = (const int*)d_in[1];
  const float* Wl1 = (const float*)d_in[2];
  const float* bl1 = (const float*)d_in[3];
  const float* Wr1 = (const float*)d_in[4];
  const float* Wl2 = (const float*)d_in[5];
  const float* bl2 = (const float*)d_in[6];
  const float* Wr2 = (const float*)d_in[7];
  const int* src = ei;
  const int* dst = ei + N_EDGES;

  float* agg    = (float*)d_ws;                         // 100000*128 f32
  float* h      = agg + (size_t)N_NODES * DIM_HID;      // 100000*128 f32
  float* deg    = h + (size_t)N_NODES * DIM_HID;        // 100000 f32
  float* colsum = deg + N_NODES;                        // 128 f32
  float* outp   = (float*)d_out;                        // 100000*64 f32

  // zero agg | h | deg | colsum (contiguous span)
  size_t zn = 2 * (size_t)N_NODES * DIM_HID + N_NODES + 128;
  zero_f32<<<4096, 256, 0, stream>>>(agg, zn);

  // ---- layer 1
  scatter_edges<<<N_EDGES / 8, 256, 0, stream>>>(x, src, dst, agg, deg, N_EDGES);
  sage_gemm<DIM_IN, DIM_HID, 8><<<N_NODES / 16, 256, 0, stream>>>(
      agg, deg, x, Wl1, bl1, Wr1, h);

  // ---- PairNorm-SCS + ReLU
  col_sum<<<625, 256, 0, stream>>>(h, colsum, N_NODES);
  pairnorm_relu<<<N_NODES / 8, 256, 0, stream>>>(h, colsum, N_NODES);

  // ---- layer 2 (deg unchanged; reuse)
  zero_f32<<<4096, 256, 0, stream>>>(agg, (size_t)N_NODES * DIM_HID);
  scatter_edges<<<N_EDGES / 8, 256, 0, stream>>>(h, src, dst, agg, nullptr, N_EDGES);
  sage_gemm<DIM_HID, DIM_OUT, 4><<<N_NODES / 32, 256, 0, stream>>>(
      agg, deg, h, Wl2, bl2, Wr2, outp);
}